// GaussianModel_14027363189459
// MI455X (gfx1250) — compile-verified
//
#include <hip/hip_runtime.h>
#include <cstdint>

#define NT  256   // threads per block
#define PPB 512   // points per block (2 per thread)

// ---- CDNA5 async global <-> LDS helpers (tracked by ASYNCcnt) ----
__device__ __forceinline__ void async_ld_b128(unsigned lds, unsigned voff, const void* sbase) {
  asm volatile("global_load_async_to_lds_b128 %0, %1, %2"
               :: "v"(lds), "v"(voff), "s"(sbase) : "memory");
}
__device__ __forceinline__ void async_ld_b32(unsigned lds, unsigned voff, const void* sbase) {
  asm volatile("global_load_async_to_lds_b32 %0, %1, %2"
               :: "v"(lds), "v"(voff), "s"(sbase) : "memory");
}
__device__ __forceinline__ void async_st_b128(unsigned voff, unsigned lds, const void* sbase) {
  asm volatile("global_store_async_from_lds_b128 %0, %1, %2"
               :: "v"(voff), "v"(lds), "s"(sbase) : "memory");
}
__device__ __forceinline__ void async_st_b32(unsigned voff, unsigned lds, const void* sbase) {
  asm volatile("global_store_async_from_lds_b32 %0, %1, %2"
               :: "v"(voff), "v"(lds), "s"(sbase) : "memory");
}
__device__ __forceinline__ void wait_async0() {
  asm volatile("s_wait_asynccnt 0x0" ::: "memory");
}

__global__ __launch_bounds__(NT)
void gauss_covar_kernel(const float* __restrict__ quats,
                        const float* __restrict__ log_scales,
                        float* __restrict__ out, int n) {
  // LDS layout (floats): quats [0, PPB*4) | scales [PPB*4, PPB*7) | covars [PPB*7, PPB*16)
  __shared__ __align__(16) float sh[PPB * 16];  // 32 KB
  const int tid = threadIdx.x;
  const int i0  = blockIdx.x * PPB;
  const int cnt = min(PPB, n - i0);

  const unsigned lds0 = (unsigned)(uintptr_t)(&sh[0]);
  const unsigned ldsQ = lds0;                // PPB*16 bytes
  const unsigned ldsS = lds0 + PPB * 16u;    // PPB*12 bytes
  const unsigned ldsO = lds0 + PPB * 28u;    // PPB*36 bytes

  // ---- async stage: global -> LDS (logit_opacities is dead: never read) ----
  const float* qb = quats      + (size_t)i0 * 4;   // 16*i0  -> 16B aligned
  const float* sb = log_scales + (size_t)i0 * 3;   // 12*i0  -> 16B aligned (i0 % 4 == 0)
  for (int t = tid; t < cnt; t += NT)              // quats: one b128 per point
    async_ld_b128(ldsQ + (unsigned)t * 16u, (unsigned)t * 16u, qb);
  {
    const int sbytes = cnt * 12;
    const int nv = sbytes >> 4;                    // dense b128 chunks
    for (int j = tid; j < nv; j += NT)
      async_ld_b128(ldsS + (unsigned)j * 16u, (unsigned)j * 16u, sb);
    for (int j = (nv << 2) + tid; j < (sbytes >> 2); j += NT)   // dword remainder
      async_ld_b32(ldsS + (unsigned)j * 4u, (unsigned)j * 4u, sb);
  }
  wait_async0();      // this wave's async LDS writes complete
  __syncthreads();    // visible to all waves

  // ---- compute: Sigma = R * diag(exp(2*log_s)) * R^T  (2 points per thread) ----
  for (int t = tid; t < cnt; t += NT) {
    float w = sh[t * 4 + 0], x = sh[t * 4 + 1];
    float y = sh[t * 4 + 2], z = sh[t * 4 + 3];
    float inv = rsqrtf(w * w + x * x + y * y + z * z);
    w *= inv; x *= inv; y *= inv; z *= inv;

    float R00 = 1.f - 2.f * (y * y + z * z);
    float R01 = 2.f * (x * y - w * z);
    float R02 = 2.f * (x * z + w * y);
    float R10 = 2.f * (x * y + w * z);
    float R11 = 1.f - 2.f * (x * x + z * z);
    float R12 = 2.f * (y * z - w * x);
    float R20 = 2.f * (x * z - w * y);
    float R21 = 2.f * (y * z + w * x);
    float R22 = 1.f - 2.f * (x * x + y * y);

    const float* ss = &sh[PPB * 4];
    float a = __expf(2.f * ss[t * 3 + 0]);   // s_x^2
    float b = __expf(2.f * ss[t * 3 + 1]);   // s_y^2
    float c = __expf(2.f * ss[t * 3 + 2]);   // s_z^2

    float* o = &sh[PPB * 7 + t * 9];
    float s00 = a * R00 * R00 + b * R01 * R01 + c * R02 * R02;
    float s01 = a * R00 * R10 + b * R01 * R11 + c * R02 * R12;
    float s02 = a * R00 * R20 + b * R01 * R21 + c * R02 * R22;
    float s11 = a * R10 * R10 + b * R11 * R11 + c * R12 * R12;
    float s12 = a * R10 * R20 + b * R11 * R21 + c * R12 * R22;
    float s22 = a * R20 * R20 + b * R21 * R21 + c * R22 * R22;
    o[0] = s00; o[1] = s01; o[2] = s02;
    o[3] = s01; o[4] = s11; o[5] = s12;
    o[6] = s02; o[7] = s12; o[8] = s22;
  }
  __syncthreads();    // result tile complete in LDS (DScnt drained by barrier)

  // ---- async drain: LDS -> global, dense 128-bit wave-coalesced stores ----
  {
    float* ob = out + (size_t)i0 * 9;              // 36*i0 -> 16B aligned
    const int obytes = cnt * 36;
    const int nov = obytes >> 4;
    for (int j = tid; j < nov; j += NT)
      async_st_b128((unsigned)j * 16u, ldsO + (unsigned)j * 16u, ob);
    for (int j = (nov << 2) + tid; j < (obytes >> 2); j += NT)
      async_st_b32((unsigned)j * 4u, ldsO + (unsigned)j * 4u, ob);
  }
  wait_async0();      // stores drained from LDS before wave retires
}

extern "C" void kernel_launch(void* const* d_in, const int* in_sizes, int n_in,
                              void* d_out, int out_size, void* d_ws, size_t ws_size,
                              hipStream_t stream) {
  const float* quats = (const float*)d_in[0];       // [N,4]
  const float* logs  = (const float*)d_in[1];       // [N,3]
  // d_in[2] (logit_opacities) intentionally unused: its sigmoid is not returned.
  float* out = (float*)d_out;                       // [N,3,3]
  const int n = in_sizes[0] / 4;
  if (n <= 0) return;
  const int grid = (n + PPB - 1) / PPB;
  gauss_covar_kernel<<<dim3(grid), dim3(NT), 0, stream>>>(quats, logs, out, n);
}